// GNNEncoder_8048768712836
// MI455X (gfx1250) — compile-verified
//
#include <hip/hip_runtime.h>

#define IN_C  128
#define H_C   128
#define OUT_C 64

typedef float v2f __attribute__((ext_vector_type(2)));
typedef float v8f __attribute__((ext_vector_type(8)));

// Non-returning hardware float atomic: global_atomic_add_f32 (STOREcnt).
// s_endpgm performs an implicit wait-idle, so no explicit s_wait needed.
__device__ __forceinline__ void atomic_fadd(float* p, float v) {
    asm volatile("global_atomic_add_f32 %0, %1, off" : : "v"(p), "v"(v) : "memory");
}

// ---------------------------------------------------------------- utilities
__global__ void fill_kernel(float* __restrict__ p, float v, int n) {
    int i = blockIdx.x * blockDim.x + threadIdx.x;
    if (i < n) p[i] = v;
}

__global__ void deg_scatter_kernel(const int* __restrict__ dst,
                                   float* __restrict__ deg, int e) {
    int i = blockIdx.x * blockDim.x + threadIdx.x;
    if (i < e) atomic_fadd(&deg[dst[i]], 1.0f);
}

__global__ void rsqrt_kernel(float* __restrict__ p, int n) {
    int i = blockIdx.x * blockDim.x + threadIdx.x;
    if (i < n) p[i] = rsqrtf(p[i]);   // deg >= 1 always (self loop)
}

// ------------------------------------------------------------ WMMA f32 GEMM
// C[M x NC] = A[M x K] * B[K x NC] (+ bias). Persistent blocks: each wave
// preloads its 16-column B fragment into VGPRs once, then grid-strides over
// 16-row strips of A staged through LDS. Inner loop: 2 ds_load + 1 v_wmma.
template<int K, int NC, bool ADD_BIAS>
__global__ void gemm_wmma_kernel(const float* __restrict__ A,
                                 const float* __restrict__ B,
                                 const float* __restrict__ bias,
                                 float* __restrict__ C, int M) {
    __shared__ __align__(16) float As[16 * K];

    const int tid   = threadIdx.x;
    const int wave  = tid >> 5;
    const int lane  = tid & 31;
    const int n0    = wave * 16;
    const int lhalf = lane >> 4;   // 0: K pair {k0,k0+1}, 1: {k0+2,k0+3}
    const int l15   = lane & 15;
    const int n     = n0 + l15;

    // B 4x16 f32 layout: row striped across lanes within a VGPR.
    // Hold all K/4 fragments of this wave's column tile in registers.
    v2f bfrag[K / 4];
#pragma unroll
    for (int k0 = 0; k0 < K; k0 += 4) {
        const int kk = k0 + 2 * lhalf;
        bfrag[k0 / 4].x = B[(size_t)kk * NC + n];
        bfrag[k0 / 4].y = B[(size_t)(kk + 1) * NC + n];
    }
    const float bv = ADD_BIAS ? bias[n] : 0.0f;

    const int strips      = (M + 15) / 16;
    const int vecsPerRow  = K / 4;

    for (int s = blockIdx.x; s < strips; s += gridDim.x) {
        const int row0 = s * 16;
        __syncthreads();   // previous strip's readers done before re-staging As
        for (int v = tid; v < 16 * vecsPerRow; v += blockDim.x) {
            int r = v / vecsPerRow;
            float4 val = make_float4(0.f, 0.f, 0.f, 0.f);
            if (row0 + r < M)
                val = ((const float4*)(A + (size_t)(row0 + r) * K))[v % vecsPerRow];
            ((float4*)As)[v] = val;
        }
        __syncthreads();

        v8f acc = {};
#pragma unroll
        for (int k0 = 0; k0 < K; k0 += 4) {
            const int kk = k0 + 2 * lhalf;
            v2f a;
            // A 16x4 f32 layout: lanes 0-15 K={k0,k0+1}, lanes 16-31 K={k0+2,k0+3}
            a.x = As[l15 * K + kk];
            a.y = As[l15 * K + kk + 1];
            acc = __builtin_amdgcn_wmma_f32_16x16x4_f32(
                      false, a, false, bfrag[k0 / 4], (short)0, acc, false, false);
        }

        // C/D layout: VGPR r -> M = r (lanes 0-15) or r+8 (lanes 16-31)
#pragma unroll
        for (int r = 0; r < 8; ++r) {
            int m = row0 + r + 8 * lhalf;
            if (m < M) C[(size_t)m * NC + n] = acc[r] + bv;
        }
    }
}

// -------------------------------------------------- GCN aggregation kernels
// out[i,c] = tmp[i,c]*dinv[i]^2 + bias[c]   (self-loop term + bias)
__global__ void agg_init_kernel(const float* __restrict__ tmp,
                                const float* __restrict__ dinv,
                                const float* __restrict__ bias,
                                float* __restrict__ out, int total) {
    int i = blockIdx.x * blockDim.x + threadIdx.x;
    if (i >= total) return;
    int node = i >> 7;            // H_C == 128
    int c    = i & 127;
    float di = dinv[node];
    out[i] = tmp[i] * di * di + bias[c];
}

// one wave32 per edge: lane l handles channels [4l, 4l+4)
__global__ void edge_scatter_kernel(const int* __restrict__ src,
                                    const int* __restrict__ dst,
                                    const float* __restrict__ dinv,
                                    const float* __restrict__ msg,
                                    float* __restrict__ out, int e) {
    int gid  = blockIdx.x * blockDim.x + threadIdx.x;
    int edge = gid >> 5;
    int lane = gid & 31;
    if (edge >= e) return;
    int s = src[edge], d = dst[edge];
    float w = dinv[s] * dinv[d];
    float4 v = ((const float4*)(msg + (size_t)s * H_C))[lane];
    float* o = out + (size_t)d * H_C + lane * 4;
    atomic_fadd(o + 0, v.x * w);
    atomic_fadd(o + 1, v.y * w);
    atomic_fadd(o + 2, v.z * w);
    atomic_fadd(o + 3, v.w * w);
}

__global__ void relu_kernel(float* __restrict__ p, int n) {
    int i = blockIdx.x * blockDim.x + threadIdx.x;
    if (i < n) p[i] = fmaxf(p[i], 0.0f);
}

// --------------------------------------------------------------- mean pool
__global__ void pool_scatter_kernel(const float* __restrict__ h,
                                    const int* __restrict__ batch,
                                    float* __restrict__ sums,
                                    float* __restrict__ cnts, int n) {
    int gid  = blockIdx.x * blockDim.x + threadIdx.x;
    int node = gid >> 5;
    int lane = gid & 31;
    if (node >= n) return;
    int g = batch[node];
    float4 v = ((const float4*)(h + (size_t)node * H_C))[lane];
    float* o = sums + (size_t)g * H_C + lane * 4;
    atomic_fadd(o + 0, v.x);
    atomic_fadd(o + 1, v.y);
    atomic_fadd(o + 2, v.z);
    atomic_fadd(o + 3, v.w);
    if (lane == 0) atomic_fadd(&cnts[g], 1.0f);
}

__global__ void pool_div_kernel(float* __restrict__ pooled,
                                const float* __restrict__ cnts, int total) {
    int i = blockIdx.x * blockDim.x + threadIdx.x;
    if (i >= total) return;
    pooled[i] /= fmaxf(cnts[i >> 7], 1.0f);
}

// ------------------------------------------------------------------ launch
extern "C" void kernel_launch(void* const* d_in, const int* in_sizes, int n_in,
                              void* d_out, int out_size, void* d_ws, size_t ws_size,
                              hipStream_t stream) {
    const float* x    = (const float*)d_in[0];
    const int*   ei   = (const int*)  d_in[1];   // [2, E]: row0 = src, row1 = dst
    const int*   batch= (const int*)  d_in[2];
    const float* W1   = (const float*)d_in[3];
    const float* b1   = (const float*)d_in[4];
    const float* W2   = (const float*)d_in[5];
    const float* b2   = (const float*)d_in[6];
    const float* Wl   = (const float*)d_in[7];
    const float* bl   = (const float*)d_in[8];

    const int N = in_sizes[0] / IN_C;
    const int E = in_sizes[1] / 2;
    const int G = (out_size - N * OUT_C) / H_C;

    const int* esrc = ei;
    const int* edst = ei + E;

    float* logits = (float*)d_out;                   // [N, OUT_C]
    float* pooled = logits + (size_t)N * OUT_C;      // [G, H_C]

    // workspace layout
    float* deg  = (float*)d_ws;                      // N (becomes dinv)
    float* tmpA = deg + (((size_t)N + 255) & ~(size_t)255);  // N*H (X@W)
    float* tmpB = tmpA + (size_t)N * H_C;            // N*H (aggregated h)
    float* cnts = tmpB + (size_t)N * H_C;            // G

    const int T  = 256;
    const int NH = N * H_C;
    const int rowBlocks  = (N + 15) / 16;
    const int gemmBlocks = rowBlocks < 512 ? rowBlocks : 512;   // persistent grid

    // degree / normalization
    fill_kernel<<<(N + T - 1) / T, T, 0, stream>>>(deg, 1.0f, N);   // self loops
    fill_kernel<<<(G + T - 1) / T, T, 0, stream>>>(cnts, 0.0f, G);
    fill_kernel<<<(G * H_C + T - 1) / T, T, 0, stream>>>(pooled, 0.0f, G * H_C);
    deg_scatter_kernel<<<(E + T - 1) / T, T, 0, stream>>>(edst, deg, E);
    rsqrt_kernel<<<(N + T - 1) / T, T, 0, stream>>>(deg, N);        // deg -> dinv

    // ---- layer 1: h1 = relu(agg(x @ W1) + b1)
    gemm_wmma_kernel<IN_C, H_C, false><<<gemmBlocks, 256, 0, stream>>>(x, W1, nullptr, tmpA, N);
    agg_init_kernel<<<(NH + T - 1) / T, T, 0, stream>>>(tmpA, deg, b1, tmpB, NH);
    edge_scatter_kernel<<<((size_t)E * 32 + T - 1) / T, T, 0, stream>>>(esrc, edst, deg, tmpA, tmpB, E);
    relu_kernel<<<(NH + T - 1) / T, T, 0, stream>>>(tmpB, NH);

    // ---- layer 2: h2 = relu(agg(h1 @ W2) + b2)
    gemm_wmma_kernel<H_C, H_C, false><<<gemmBlocks, 256, 0, stream>>>(tmpB, W2, nullptr, tmpA, N);
    agg_init_kernel<<<(NH + T - 1) / T, T, 0, stream>>>(tmpA, deg, b2, tmpB, NH);
    edge_scatter_kernel<<<((size_t)E * 32 + T - 1) / T, T, 0, stream>>>(esrc, edst, deg, tmpA, tmpB, E);
    relu_kernel<<<(NH + T - 1) / T, T, 0, stream>>>(tmpB, NH);

    // ---- logits = h2 @ Wl + bl
    gemm_wmma_kernel<H_C, OUT_C, true><<<gemmBlocks, 128, 0, stream>>>(tmpB, Wl, bl, logits, N);

    // ---- pooled = segment_mean(h2, batch)
    pool_scatter_kernel<<<((size_t)N * 32 + T - 1) / T, T, 0, stream>>>(tmpB, batch, pooled, cnts, N);
    pool_div_kernel<<<(G * H_C + T - 1) / T, T, 0, stream>>>(pooled, cnts, G * H_C);
}